// Exprnn_2542620639651
// MI455X (gfx1250) — compile-verified
//
#include <hip/hip_runtime.h>

// ExpRNN fused forward for MI455X (gfx1250, wave32).
// Transposed formulation: activations live as x^T in WMMA D-layout (feature
// over VGPRs, batch over lanes); weights preloaded as W^T in WMMA A-layout;
// every layer = V_WMMA_F32_16X16X4_F32 with bias preloaded into C.
// Decoder folded: out = h @ (W3@W4) + (b3@W4 + b4)  (no activation between).
// Elementwise ops (relu / modrelu) are applied in B-fragment space (6 comps)
// after the layout conversion; the recurrent state is kept ONLY as B-frags.
// Recurrent matmul uses 3 parallel partial accumulators to shorten the
// loop-carried dependence chain (latency-bound: 512 serial steps/wave).

typedef float v2f __attribute__((ext_vector_type(2)));
typedef float v8f __attribute__((ext_vector_type(8)));

#define HDIM 10
#define TSTEPS 512

// relu in one VALU op (inputs are never NaN here)
__device__ __forceinline__ float relu1(float x) {
    return __builtin_amdgcn_fmed3f(x, 0.0f, __builtin_inff());
}

__device__ __forceinline__ float swpf(int swpaddr, float v) {
    // value of v at lane ^ 16 (half-wave swap)
    return __int_as_float(__builtin_amdgcn_ds_bpermute(swpaddr, __float_as_int(v)));
}

__device__ __forceinline__ v8f wmma4(v2f a, v2f b, v8f c) {
    // D(16x16,f32) = A(16x4,f32) x B(4x16,f32) + C
    return __builtin_amdgcn_wmma_f32_16x16x4_f32(
        /*neg_a=*/false, a, /*neg_b=*/false, b,
        /*c_mod=*/(short)0, c, /*reuse_a=*/false, /*reuse_b=*/false);
}

// W^T slice q in A-layout. W row-major [kdim, ndim] (y = x @ W).
// lanes 0-15: M=lane, VGPR p -> K=4q+p ; lanes 16-31: M=lane-16, K=4q+2+p.
__device__ __forceinline__ v2f wfrag(const float* __restrict__ W,
                                     int kdim, int ndim, int q, int lane) {
    const int m = lane & 15;
    const int half = lane >> 4;
    v2f a;
#pragma unroll
    for (int p = 0; p < 2; ++p) {
        const int k = 4 * q + 2 * half + p;
        a[p] = (m < ndim && k < kdim) ? W[k * ndim + m] : 0.0f;
    }
    return a;
}

// (W3@W4)^T slice q in A-layout: element [m][k] = sum_j W3[k][j] * W4[j][m].
__device__ __forceinline__ v2f w34frag(const float* __restrict__ W3,
                                       const float* __restrict__ W4,
                                       int q, int lane) {
    const int m = lane & 15;
    const int half = lane >> 4;
    v2f a;
#pragma unroll
    for (int p = 0; p < 2; ++p) {
        const int k = 4 * q + 2 * half + p;
        float s = 0.0f;
        if (m < HDIM && k < HDIM) {
            for (int j = 0; j < HDIM; ++j)
                s += W3[k * HDIM + j] * W4[j * HDIM + m];
        }
        a[p] = s;
    }
    return a;
}

// Length-10 vector broadcast into D-layout C (rows over VGPR/half structure).
__device__ __forceinline__ v8f biasfrag(const float* __restrict__ b, int lane) {
    const int off = (lane < 16) ? 0 : 8;
    v8f c;
#pragma unroll
    for (int v = 0; v < 8; ++v) {
        const int j = v + off;
        c[v] = (j < HDIM) ? b[j] : 0.0f;
    }
    return c;
}

// Length-10 vector in B-fragment k-indexing: slice q, comp p -> k=4q+2*half+p.
__device__ __forceinline__ v2f kvecfrag(const float* __restrict__ b,
                                        int q, int lane) {
    const int half = lane >> 4;
    v2f a;
#pragma unroll
    for (int p = 0; p < 2; ++p) {
        const int k = 4 * q + 2 * half + p;
        a[p] = (k < HDIM) ? b[k] : 0.0f;
    }
    return a;
}

// Folded decoder bias: b34[j] = b4[j] + sum_i b3[i] * W4[i][j], in D-layout.
__device__ __forceinline__ v8f bias34frag(const float* __restrict__ b3,
                                          const float* __restrict__ W4,
                                          const float* __restrict__ b4, int lane) {
    const int off = (lane < 16) ? 0 : 8;
    v8f c;
#pragma unroll
    for (int v = 0; v < 8; ++v) {
        const int j = v + off;
        float s = 0.0f;
        if (j < HDIM) {
            s = b4[j];
            for (int i = 0; i < HDIM; ++i) s += b3[i] * W4[i * HDIM + j];
        }
        c[v] = s;
    }
    return c;
}

struct Frags { v2f q[3]; };

// D-layout (rows 0..9 valid) -> three B-layout K-slices.
// Only cross-half rows need a shuffle: 6 ds_bpermute + 6 cndmask.
//  slice0: rows 0,1 (own, lo) / 2,3 (swapped, hi)
//  slice1: rows 4,5 (own, lo) / 6,7 (swapped, hi)
//  slice2: rows 8,9 (swapped into lo) / rows 10,11 == 0
__device__ __forceinline__ Frags d_to_b(const v8f d, bool lo, int swpaddr) {
    const float s0 = swpf(swpaddr, d[0]);
    const float s1 = swpf(swpaddr, d[1]);
    const float s2 = swpf(swpaddr, d[2]);
    const float s3 = swpf(swpaddr, d[3]);
    const float s6 = swpf(swpaddr, d[6]);
    const float s7 = swpf(swpaddr, d[7]);
    Frags f;
    f.q[0][0] = lo ? d[0] : s2;
    f.q[0][1] = lo ? d[1] : s3;
    f.q[1][0] = lo ? d[4] : s6;
    f.q[1][1] = lo ? d[5] : s7;
    f.q[2][0] = lo ? s0 : 0.0f;
    f.q[2][1] = lo ? s1 : 0.0f;
    return f;
}

// relu applied in B-fragment space (6 components; zero padding stays zero)
__device__ __forceinline__ void relu_frags(Frags& f) {
#pragma unroll
    for (int i = 0; i < 3; ++i) {
        f.q[i][0] = relu1(f.q[i][0]);
        f.q[i][1] = relu1(f.q[i][1]);
    }
}

__global__ void __launch_bounds__(256)
exprnn_fused_kernel(const float* __restrict__ X,
                    const float* __restrict__ W1, const float* __restrict__ b1,
                    const float* __restrict__ W2, const float* __restrict__ b2,
                    const float* __restrict__ Win, const float* __restrict__ R,
                    const float* __restrict__ mb,
                    const float* __restrict__ W3, const float* __restrict__ b3,
                    const float* __restrict__ W4, const float* __restrict__ b4,
                    float* __restrict__ out, int Bsz) {
    const int lane = threadIdx.x & 31;
    const int wid = (blockIdx.x * blockDim.x + threadIdx.x) >> 5;
    const int b0 = wid * 16;                 // 16 batch rows per wave
    if (b0 >= Bsz) return;                   // wave-uniform exit

    const int sl = lane & 15;
    const bool lo = lane < 16;
    const int swpaddr = (lane ^ 16) << 2;
    const int brow = b0 + sl;
    const bool rowok = brow < Bsz;
    const int crow = rowok ? brow : (Bsz - 1);   // clamp for branchless loads

    // ---- preload weights (A-layout W^T fragments) and biases ----
    const v2f a1 = wfrag(W1, 2, HDIM, 0, lane);  // K=2 fits one slice
    v2f a2[3], au[3], ar[3], a34[3], mbb[3];
#pragma unroll
    for (int q = 0; q < 3; ++q) {
        a2[q]  = wfrag(W2, HDIM, HDIM, q, lane);
        au[q]  = wfrag(Win, HDIM, HDIM, q, lane);
        ar[q]  = wfrag(R, HDIM, HDIM, q, lane);
        a34[q] = w34frag(W3, W4, q, lane);
        mbb[q] = kvecfrag(mb, q, lane);          // modrelu bias, B k-indexing
    }
    const v8f c1  = biasfrag(b1, lane);
    const v8f c2  = biasfrag(b2, lane);
    const v8f c34 = bias34frag(b3, W4, b4, lane);
    const v8f zf = {0.0f, 0.0f, 0.0f, 0.0f, 0.0f, 0.0f, 0.0f, 0.0f};

    // recurrent state lives ONLY as B-fragments (h^T, K-sliced); h0 = 0
    Frags hb;
#pragma unroll
    for (int i = 0; i < 3; ++i) hb.q[i] = (v2f){0.0f, 0.0f};

    // Lane row bases. X: [B, T, 2]; out: [B, T, 10].
    const float* xp = X + (size_t)crow * TSTEPS * 2;
    float* op = out + (size_t)brow * TSTEPS * 10 + (lo ? 0 : 8);

    auto step = [&](v2f bx, int t) {
        // encoder fc1_1: W1^T x^T + b1  (relu applied in B space below)
        v8f d = wmma4(a1, bx, c1);
        Frags f = d_to_b(d, lo, swpaddr);
        relu_frags(f);
        // encoder fc1_2: W2^T (.) + b2
        d = c2;
        d = wmma4(a2[0], f.q[0], d);
        d = wmma4(a2[1], f.q[1], d);
        d = wmma4(a2[2], f.q[2], d);
        f = d_to_b(d, lo, swpaddr);
        relu_frags(f);
        // input kernel: u^T = Win^T (.)  (bias-free)
        v8f u = wmma4(au[0], f.q[0], zf);
        u = wmma4(au[1], f.q[1], u);
        u = wmma4(au[2], f.q[2], u);
        // recurrence: z = u + R^T h^T  (3 parallel partials, 2 vector adds)
        const v8f z0 = wmma4(ar[0], hb.q[0], u);
        const v8f z1 = wmma4(ar[1], hb.q[1], zf);
        const v8f z2 = wmma4(ar[2], hb.q[2], zf);
        const v8f z = (z0 + z1) + z2;
        // convert z, then modrelu directly in B space -> new recurrent state
        Frags zb = d_to_b(z, lo, swpaddr);
#pragma unroll
        for (int i = 0; i < 3; ++i) {
#pragma unroll
            for (int p = 0; p < 2; ++p) {
                const float zz = zb.q[i][p];
                const float m = relu1(fabsf(zz) + mbb[i][p]);
                zb.q[i][p] = (zz > 0.0f) ? m : ((zz < 0.0f) ? -m : 0.0f);
            }
        }
        hb = zb;   // feeds folded decoder now AND recurrence next step
        // folded decoder: out^T = (W3W4)^T h^T + b34
        v8f y = c34;
        y = wmma4(a34[0], hb.q[0], y);
        y = wmma4(a34[1], hb.q[1], y);
        y = wmma4(a34[2], hb.q[2], y);
        // store: lanes 0-15 cols 0..7, lanes 16-31 cols 8..9 (8B aligned)
        if (rowok) {
            float* o = op + (size_t)t * 10;
            if (lo) {
                *(float2*)(o + 0) = make_float2(y[0], y[1]);
                *(float2*)(o + 2) = make_float2(y[2], y[3]);
                *(float2*)(o + 4) = make_float2(y[4], y[5]);
                *(float2*)(o + 6) = make_float2(y[6], y[7]);
            } else {
                *(float2*)(o) = make_float2(y[0], y[1]);
            }
        }
    };

    for (int t = 0; t < TSTEPS; t += 2) {
        // one aligned float4 load covers two timesteps of this lane's row
        const float4 f4 = *(const float4*)(xp + (size_t)t * 2);
        v2f bx;
        bx[0] = lo ? f4.x : 0.0f;   // B rows K=0,1 live in lanes 0-15
        bx[1] = lo ? f4.y : 0.0f;   // rows K=2,3 (lanes 16-31) are zero pad
        step(bx, t);
        bx[0] = lo ? f4.z : 0.0f;
        bx[1] = lo ? f4.w : 0.0f;
        step(bx, t + 1);
    }
}

extern "C" void kernel_launch(void* const* d_in, const int* in_sizes, int n_in,
                              void* d_out, int out_size, void* d_ws, size_t ws_size,
                              hipStream_t stream) {
    const float* X   = (const float*)d_in[0];
    const float* W1  = (const float*)d_in[1];
    const float* b1  = (const float*)d_in[2];
    const float* W2  = (const float*)d_in[3];
    const float* b2  = (const float*)d_in[4];
    const float* Win = (const float*)d_in[5];
    const float* R   = (const float*)d_in[6];
    const float* mb  = (const float*)d_in[7];
    const float* W3  = (const float*)d_in[8];
    const float* b3  = (const float*)d_in[9];
    const float* W4  = (const float*)d_in[10];
    const float* b4  = (const float*)d_in[11];
    float* out = (float*)d_out;

    const int BT = out_size / 10;      // B*T
    const int B = BT / TSTEPS;         // reference setup: T = 512
    if (B <= 0) return;

    const int tiles = (B + 15) / 16;   // one wave32 per 16 batch rows
    const int wavesPerBlock = 8;       // 256 threads
    const int blocks = (tiles + wavesPerBlock - 1) / wavesPerBlock;

    exprnn_fused_kernel<<<blocks, wavesPerBlock * 32, 0, stream>>>(
        X, W1, b1, W2, b2, Win, R, mb, W3, b3, W4, b4, out, B);
}